// WordCountingModule_26010321944672
// MI455X (gfx1250) — compile-verified
//
#include <hip/hip_runtime.h>
#include <math.h>

// ---------------------------------------------------------------------------
// Problem constants (from reference): B=1024, A=4, V=32000. B/A derived from
// in_sizes at launch; V fixed by the reference model.
// ---------------------------------------------------------------------------
#define VDIM 32000
#define BLK  256           // 8 wave32 waves per block

// gfx1250 async global->LDS path (guarded: falls back to plain b128 loads)
#if defined(__has_builtin)
#  if __has_builtin(__builtin_amdgcn_global_load_async_to_lds_b128) && \
      __has_builtin(__builtin_amdgcn_s_wait_asynccnt)
#    define USE_ASYNC_LDS 1
#  endif
#endif

typedef int v4i __attribute__((ext_vector_type(4)));
typedef __attribute__((address_space(1))) v4i as1_v4i;
typedef __attribute__((address_space(3))) v4i as3_v4i;

// ---------------------------------------------------------------------------
// Kernel A: per-(b,a) row argmax over V. One 256-thread block per row.
// float4 (b128) streaming loads + global_prefetch_b8 hints.
// Tie-break: first (lowest) index, matching jnp.argmax.
// ---------------------------------------------------------------------------
__global__ void __launch_bounds__(BLK)
argmax_rows_kernel(const float* __restrict__ utt, int* __restrict__ idx_out)
{
    const int row = blockIdx.x;
    const int tid = threadIdx.x;
    const float4* __restrict__ p4 =
        (const float4*)(utt + (size_t)row * VDIM);
    const int nv4 = VDIM >> 2;      // 8000

    float best    = -INFINITY;
    int   bestIdx = 0x7fffffff;

    for (int i = tid; i < nv4; i += BLK) {
        // prefetch two strides ahead (speculative; OOB is silently dropped)
        __builtin_prefetch((const void*)(p4 + i + 2 * BLK), 0, 0);
        float4 v = p4[i];
        int base = i << 2;
        if (v.x > best) { best = v.x; bestIdx = base;     }
        if (v.y > best) { best = v.y; bestIdx = base + 1; }
        if (v.z > best) { best = v.z; bestIdx = base + 2; }
        if (v.w > best) { best = v.w; bestIdx = base + 3; }
    }

    // wave32 shuffle reduction (argmax, prefer lower index on tie)
    for (int off = 16; off > 0; off >>= 1) {
        float ov = __shfl_down(best,    (unsigned)off, 32);
        int   oi = __shfl_down(bestIdx, (unsigned)off, 32);
        if (ov > best || (ov == best && oi < bestIdx)) { best = ov; bestIdx = oi; }
    }

    __shared__ float sv[BLK / 32];
    __shared__ int   si[BLK / 32];
    const int lane = tid & 31;
    const int wave = tid >> 5;
    if (lane == 0) { sv[wave] = best; si[wave] = bestIdx; }
    __syncthreads();

    if (tid == 0) {
        for (int w = 1; w < BLK / 32; ++w) {
            if (sv[w] > best || (sv[w] == best && si[w] < bestIdx)) {
                best = sv[w]; bestIdx = si[w];
            }
        }
        idx_out[row] = bestIdx;
    }
}

// ---------------------------------------------------------------------------
// Kernel B: per-batch row. Streams word_counts with a double-buffered async
// global->LDS (ASYNCcnt) pipeline: chunk c+1 is in flight while chunk c is
// consumed (wait asynccnt<=1). Adds indicator hits for the <=4 chosen words,
// divides by the timestep-derived denominator, writes prob_ck. Thread 0 also
// computes the per-batch sum of log-probs at the chosen indices.
// ---------------------------------------------------------------------------
__device__ __forceinline__ void process4(float* __restrict__ orow, int v0,
                                         float4 c, float denom,
                                         int i0, int i1, int i2, int i3)
{
    orow[v0    ] = (c.x + (float)((v0    ==i0)+(v0    ==i1)+(v0    ==i2)+(v0    ==i3))) / denom;
    orow[v0 + 1] = (c.y + (float)((v0+1==i0)+(v0+1==i1)+(v0+1==i2)+(v0+1==i3))) / denom;
    orow[v0 + 2] = (c.z + (float)((v0+2==i0)+(v0+2==i1)+(v0+2==i2)+(v0+2==i3))) / denom;
    orow[v0 + 3] = (c.w + (float)((v0+3==i0)+(v0+3==i1)+(v0+3==i2)+(v0+3==i3))) / denom;
}

__global__ void __launch_bounds__(BLK)
prob_reward_kernel(const float* __restrict__ wc,
                   const int*   __restrict__ idxs,
                   const int*   __restrict__ tstep,
                   float*       __restrict__ prob_out,   // d_out + 1
                   float*       __restrict__ partial,    // [B]
                   int A)
{
    const int b   = blockIdx.x;
    const int tid = threadIdx.x;

    __shared__ int sIdx[4];
    if (tid < 4) sIdx[tid] = (tid < A) ? idxs[b * A + tid] : -1;
    __syncthreads();

    const int i0 = sIdx[0], i1 = sIdx[1], i2 = sIdx[2], i3 = sIdx[3];
    const int   t     = *tstep;
    const float denom = 6.0f + (float)((t + 1) * A) - 1.0f;   // OOV_PROB + n - 1

    const float* __restrict__ wrow = wc       + (size_t)b * VDIM;
    float*       __restrict__ orow = prob_out + (size_t)b * VDIM;
    const int nv4 = VDIM >> 2;                 // 8000 float4s per row
    const float4* __restrict__ w4 = (const float4*)wrow;

#if defined(USE_ASYNC_LDS)
    const int FULL = nv4 / BLK;                // 31 full 256-wide chunks
    __shared__ float tile[2 * BLK * 4];        // 8 KB ping-pong staging

    // prologue: chunk 0 -> buffer 0 (all lanes in range for full chunks)
    __builtin_amdgcn_global_load_async_to_lds_b128(
        (as1_v4i*)(w4 + tid), (as3_v4i*)(tile + tid * 4), 0, 0);

    for (int c = 0; c < FULL; ++c) {
        if (c + 1 < FULL) {
            // keep next chunk's DMA in flight while consuming this one
            __builtin_amdgcn_global_load_async_to_lds_b128(
                (as1_v4i*)(w4 + (c + 1) * BLK + tid),
                (as3_v4i*)(tile + ((c + 1) & 1) * (BLK * 4) + tid * 4), 0, 0);
            __builtin_amdgcn_s_wait_asynccnt(1);   // chunk c complete
        } else {
            __builtin_amdgcn_s_wait_asynccnt(0);   // last chunk complete
        }
        const float4 v =
            *(const float4*)(tile + (c & 1) * (BLK * 4) + tid * 4);
        process4(orow, (c * BLK + tid) << 2, v, denom, i0, i1, i2, i3);
    }
    // tail: remaining nv4 - FULL*BLK float4s via direct b128 loads
    for (int i = FULL * BLK + tid; i < nv4; i += BLK) {
        process4(orow, i << 2, w4[i], denom, i0, i1, i2, i3);
    }
#else
    for (int i = tid; i < nv4; i += BLK) {
        __builtin_prefetch((const void*)(w4 + i + 2 * BLK), 0, 0);
        process4(orow, i << 2, w4[i], denom, i0, i1, i2, i3);
    }
#endif

    if (tid == 0) {
        float r = 0.0f;
        for (int a = 0; a < A; ++a) {
            const int ia = sIdx[a];
            const float cnt = (float)((ia==i0)+(ia==i1)+(ia==i2)+(ia==i3));
            const float p = (wrow[ia] + cnt) / denom;
            r += logf(p);
        }
        partial[b] = r;
    }
}

// ---------------------------------------------------------------------------
// Kernel C: deterministic reduction of per-batch partials -> d_out[0] = -reward
// ---------------------------------------------------------------------------
__global__ void __launch_bounds__(BLK)
reduce_reward_kernel(const float* __restrict__ partial, float* __restrict__ out0, int B)
{
    __shared__ float s[BLK];
    float acc = 0.0f;
    for (int i = threadIdx.x; i < B; i += BLK) acc += partial[i];
    s[threadIdx.x] = acc;
    __syncthreads();
    for (int off = BLK / 2; off > 0; off >>= 1) {
        if (threadIdx.x < (unsigned)off) s[threadIdx.x] += s[threadIdx.x + off];
        __syncthreads();
    }
    if (threadIdx.x == 0) out0[0] = -s[0];
}

// ---------------------------------------------------------------------------
// Host entry
// ---------------------------------------------------------------------------
extern "C" void kernel_launch(void* const* d_in, const int* in_sizes, int n_in,
                              void* d_out, int out_size, void* d_ws, size_t ws_size,
                              hipStream_t stream)
{
    const float* utt   = (const float*)d_in[0];  // [B, A, V]
    const float* wc    = (const float*)d_in[1];  // [B, V]
    const int*   tstep = (const int*)  d_in[2];  // scalar

    const int B = in_sizes[1] / VDIM;
    const int A = in_sizes[0] / in_sizes[1];

    float* out      = (float*)d_out;
    float* prob_out = out + 1;                   // [B, V] after the scalar

    int*   idxs    = (int*)d_ws;                              // [B*A]
    float* partial = (float*)((char*)d_ws + sizeof(int) * (size_t)B * A); // [B]

    argmax_rows_kernel<<<B * A, BLK, 0, stream>>>(utt, idxs);
    prob_reward_kernel<<<B, BLK, 0, stream>>>(wc, idxs, tstep, prob_out, partial, A);
    reduce_reward_kernel<<<1, BLK, 0, stream>>>(partial, out, B);
}